// Box_Rel_Classifier_89215060672871
// MI455X (gfx1250) — compile-verified
//
#include <hip/hip_runtime.h>
#include <math.h>

// ---------------------------------------------------------------------------
// Box_Rel_Classifier: gumbel-box intersection, pure streaming-elementwise op.
// Bandwidth bound: 335.5 MB of f32 output -> ~14.4 us floor at 23.3 TB/s.
// Strategy: precompute z tables (2.4 MB, L2-resident) once, then one
// store-bound kernel with LDS-staged (async-to-LDS) z2 rows, B128 loads and
// non-temporal B128 stores.  Fast log in the hot loop keeps the TRANS-pipe
// demand (~12 T ops/s at roofline) as low as the math allows.
// ---------------------------------------------------------------------------

#define GB      0.0036f
#define INV_GB  (1.0f / 0.0036f)
#define N1C     160
#define N2C     1024
#define DC      256

typedef float v4f __attribute__((ext_vector_type(4)));
typedef int   v4i __attribute__((ext_vector_type(4)));

// Pointer flavors for the async-to-LDS builtin:
// (as1 v4i*, as3 v4i*, imm int offset, imm int cpol)
typedef __attribute__((address_space(1))) v4i* gv4i_p;
typedef __attribute__((address_space(3))) v4i* lv4i_p;

// --- async-to-LDS path availability (gfx1250) ------------------------------
#if defined(__HIP_DEVICE_COMPILE__) &&                                  \
    __has_builtin(__builtin_amdgcn_global_load_async_to_lds_b128) &&    \
    __has_builtin(__builtin_amdgcn_s_wait_asynccnt)
#define USE_ASYNC_LDS 1
#else
#define USE_ASYNC_LDS 0
#endif

// ---------------------------------------------------------------------------
// Math helpers
// ---------------------------------------------------------------------------
__device__ __forceinline__ float sigmoidf_dev(float x) {
  // stable: for x << 0, __expf(-x) -> inf -> result 0; for x >> 0 -> 1.
  return 1.0f / (1.0f + __expf(-x));
}

__device__ __forceinline__ float softplus10(float o) {
  // softplus(10*o)/10, overflow-stable (precise log1pf: runs on 0.3M elems,
  // its accuracy propagates into all 84M outputs)
  float x = 10.0f * o;
  float sp = (x > 0.0f) ? (x + log1pf(__expf(-x))) : log1pf(__expf(x));
  return 0.1f * sp;
}

// gb * log1p(exp(-|d|/gb)) with e in (0,1]: __logf(1+e) error is <= ~1e-7
// before the 0.0036 scale -> ~4e-10 absolute in the output.  One v_exp_f32 +
// one v_log_f32 per element, no libm branches.
__device__ __forceinline__ float pair_min(float a, float b) {
  float m = fmaxf(a, b);
  float e = __expf(-fabsf(a - b) * INV_GB);
  float l = fmaf(GB, __logf(1.0f + e), m);
  return fmaxf(l, m);
}

__device__ __forceinline__ float pair_max(float A, float B) {
  float m = fminf(A, B);
  float e = __expf(-fabsf(A - B) * INV_GB);
  float l = fmaf(-GB, __logf(1.0f + e), m);
  return fminf(l, m);
}

// ---------------------------------------------------------------------------
// Kernel A: z tables.  box layout (nrows, 2, 256): [row,0,d]=center,
// [row,1,d]=offset.  zmin/zmax are (nrows, 256) row-major.
// ---------------------------------------------------------------------------
__global__ void precompute_kernel(const float* __restrict__ box,
                                  float* __restrict__ zmin,
                                  float* __restrict__ zmax, int nrows) {
  int idx = blockIdx.x * blockDim.x + threadIdx.x;
  if (idx >= nrows * DC) return;
  int row = idx >> 8;
  int d   = idx & (DC - 1);
  float c = box[row * 2 * DC + d];
  float o = box[row * 2 * DC + DC + d];
  float sp = softplus10(o);
  zmin[idx] = sigmoidf_dev(c - sp);
  zmax[idx] = sigmoidf_dev(c + sp);
}

// ---------------------------------------------------------------------------
// Kernel B: one block = one i (z2 row) x 4 consecutive j (z1 rows).
// 256 threads: jl = t>>6 selects j, (t&63)*4 selects the float4 column.
// z2 rows (shared x4 within block) staged to LDS via async-to-LDS B128.
// ---------------------------------------------------------------------------
__global__ void __launch_bounds__(256) pair_kernel(
    const float* __restrict__ z1min, const float* __restrict__ z1max,
    const float* __restrict__ z2min, const float* __restrict__ z2max,
    float* __restrict__ out_min, float* __restrict__ out_max) {
  __shared__ float s_a[DC];  // z2min row for this i
  __shared__ float s_A[DC];  // z2max row for this i

  const int t    = threadIdx.x;
  const int bid  = blockIdx.x;
  const int i    = bid / (N1C / 4);
  const int jblk = bid - i * (N1C / 4);

#if USE_ASYNC_LDS
  if (t < 64) {
    __builtin_amdgcn_global_load_async_to_lds_b128(
        (gv4i_p)(void*)(const void*)(z2min + (size_t)i * DC + t * 4),
        (lv4i_p)(s_a + t * 4),
        /*offset=*/0, /*cpol=*/0);
  } else if (t < 128) {
    __builtin_amdgcn_global_load_async_to_lds_b128(
        (gv4i_p)(void*)(const void*)(z2max + (size_t)i * DC + (t - 64) * 4),
        (lv4i_p)(s_A + (t - 64) * 4),
        /*offset=*/0, /*cpol=*/0);
  }
  __builtin_amdgcn_s_wait_asynccnt(0);
  __syncthreads();
#else
  if (t < 64) {
    *(v4f*)(s_a + t * 4) = *(const v4f*)(z2min + (size_t)i * DC + t * 4);
  } else if (t < 128) {
    *(v4f*)(s_A + (t - 64) * 4) = *(const v4f*)(z2max + (size_t)i * DC + (t - 64) * 4);
  }
  __syncthreads();
#endif

  const int jl = t >> 6;
  const int c4 = (t & 63) * 4;
  const int j  = jblk * 4 + jl;

  v4f a = *(const v4f*)(&s_a[c4]);                       // ds_load_b128
  v4f A = *(const v4f*)(&s_A[c4]);
  v4f b = *(const v4f*)(z1min + (size_t)j * DC + c4);    // global_load_b128 (L2-hot)
  v4f B = *(const v4f*)(z1max + (size_t)j * DC + c4);

  v4f om, oM;
#pragma unroll
  for (int k = 0; k < 4; ++k) {
    om[k] = pair_min(a[k], b[k]);
    oM[k] = pair_max(A[k], B[k]);
  }

  const size_t off = (((size_t)i * N1C) + j) * (size_t)DC + (size_t)c4;
  // Streaming output, written once, never re-read: non-temporal B128 stores.
  __builtin_nontemporal_store(om, (v4f*)(out_min + off));
  __builtin_nontemporal_store(oM, (v4f*)(out_max + off));
}

// ---------------------------------------------------------------------------
// Host entry
// ---------------------------------------------------------------------------
extern "C" void kernel_launch(void* const* d_in, const int* in_sizes, int n_in,
                              void* d_out, int out_size, void* d_ws, size_t ws_size,
                              hipStream_t stream) {
  (void)in_sizes; (void)n_in; (void)out_size; (void)ws_size;

  const float* box1 = (const float*)d_in[0];  // (160, 2, 256)
  const float* box2 = (const float*)d_in[1];  // (1024, 2, 256)

  // workspace layout (floats): z1min | z1max | z2min | z2max  (~2.4 MB total)
  float* ws    = (float*)d_ws;
  float* z1min = ws;
  float* z1max = z1min + (size_t)N1C * DC;
  float* z2min = z1max + (size_t)N1C * DC;
  float* z2max = z2min + (size_t)N2C * DC;

  precompute_kernel<<<(N1C * DC + 255) / 256, 256, 0, stream>>>(box1, z1min, z1max, N1C);
  precompute_kernel<<<(N2C * DC + 255) / 256, 256, 0, stream>>>(box2, z2min, z2max, N2C);

  float* out_min = (float*)d_out;
  float* out_max = out_min + (size_t)N2C * N1C * DC;

  // grid: 1024 i-rows x 40 j-quads = 40,960 blocks, 256 threads each
  pair_kernel<<<N2C * (N1C / 4), 256, 0, stream>>>(z1min, z1max, z2min, z2max,
                                                   out_min, out_max);
}